// MessageGraphConvolution_31671088841315
// MI455X (gfx1250) — compile-verified
//
#include <hip/hip_runtime.h>

// ---------------------------------------------------------------------------
// MessageGraphConvolution for MI455X (gfx1250, wave32)
//   out = (scatter_mean-ish(x[src] -> dst)) @ W^T + x @ B^T
// Phase 1: zero scratch (agg[100000x64], deg[100000])
// Phase 2: edge gather/scatter-add, one thread per (edge, feature)
// Phase 3: WMMA f32 16x16x4 dual GEMM, one wave per 16x64 output tile
// ---------------------------------------------------------------------------

#define N_NODES 100000
#define N_EDGES 1600000
#define D       64

typedef __attribute__((ext_vector_type(2))) float v2f;
typedef __attribute__((ext_vector_type(8))) float v8f;

__global__ __launch_bounds__(256) void zero_kernel(float* __restrict__ p, int n) {
    int i = blockIdx.x * blockDim.x + threadIdx.x;
    if (i < n) p[i] = 0.0f;
}

// One thread per (edge, feature). 64 consecutive threads share one edge, so
// the x-row gather is a fully coalesced 256B read and the scatter is 64
// consecutive global_atomic_add_f32 into L2-resident agg.
__global__ __launch_bounds__(256) void edge_scatter_kernel(
    const long long* __restrict__ src_idx,
    const long long* __restrict__ dst_idx,
    const float* __restrict__ x,
    float* __restrict__ agg,
    float* __restrict__ deg,
    int n_edges)
{
    long long t = (long long)blockIdx.x * blockDim.x + threadIdx.x;
    int e = (int)(t >> 6);
    int f = (int)(t & 63);
    if (e >= n_edges) return;
    int s = (int)src_idx[e];
    int d = (int)dst_idx[e];
    float v = x[(size_t)s * D + f];
    atomicAdd(&agg[(size_t)d * D + f], v);
    if (f == 0) atomicAdd(&deg[d], 1.0f);
}

// One wave per 16-row x 64-col output tile.
//   out[m,n] = sum_k aggN[m,k]*W[n,k] + sum_k x[m,k]*B[n,k]
// Full-precision f32 WMMA (V_WMMA_F32_16X16X4_F32).
// A 16x4 layout (ISA 7.12.2): lanes 0-15 -> M=lane, K=k0+{0,1};
//                             lanes 16-31 -> M=lane-16, K=k0+{2,3}.
// B 4x16 / C 16x16: rows striped across lanes within a VGPR.
__global__ __launch_bounds__(256) void gcn_wmma_kernel(
    const float* __restrict__ agg,
    const float* __restrict__ deg,
    const float* __restrict__ x,
    const float* __restrict__ Wm,   // [64,64], row n holds W[n, :]
    const float* __restrict__ Bm,   // [64,64], row n holds B[n, :]
    float* __restrict__ out,
    int n_nodes)
{
    const int lane  = threadIdx.x & 31;
    const int wave  = threadIdx.x >> 5;
    const int mtile = blockIdx.x * 8 + wave;
    if (mtile * 16 >= n_nodes) return;         // wave-uniform: EXEC stays all-ones
    const int m0    = mtile * 16;
    const int row   = m0 + (lane & 15);        // this lane's A-matrix row
    const int khalf = (lane >> 4) * 2;         // K sub-offset: 0 (lanes 0-15) / 2 (16-31)

    // degree normalization folded into A fragments (zero-degree -> 1)
    float dg   = deg[row];
    float invd = (dg == 0.0f) ? 1.0f : 1.0f / dg;

    // Preload all A fragments for both matrices: K = 0..63 in steps of 4.
    v2f a_agg[16];
    v2f a_x[16];
    const float* aggRow = agg + (size_t)row * D + khalf;
    const float* xRow   = x   + (size_t)row * D + khalf;
#pragma unroll
    for (int k = 0; k < 16; ++k) {
        v2f t = *(const v2f*)(aggRow + 4 * k);
        a_agg[k] = t * invd;
        a_x[k]   = *(const v2f*)(xRow + 4 * k);
    }

    const int ncol = lane & 15;
#pragma unroll
    for (int nt = 0; nt < 4; ++nt) {
        const int n0 = nt * 16;
        // B fragment source: Bfrag[k][n] = Weight[n0+n][k0+k] (W^T on the fly)
        const float* wCol = Wm + (size_t)(n0 + ncol) * D + khalf;
        const float* bCol = Bm + (size_t)(n0 + ncol) * D + khalf;
        v8f c = {};
#pragma unroll
        for (int k = 0; k < 16; ++k) {
            v2f bw = *(const v2f*)(wCol + 4 * k);
            c = __builtin_amdgcn_wmma_f32_16x16x4_f32(
                    false, a_agg[k], false, bw, (short)0, c, false, false);
            v2f bb = *(const v2f*)(bCol + 4 * k);
            c = __builtin_amdgcn_wmma_f32_16x16x4_f32(
                    false, a_x[k], false, bb, (short)0, c, false, false);
        }
        // C/D 16x16 layout: VGPR r -> row m0+r (lanes 0-15) / m0+8+r (lanes 16-31)
        const int rbase = m0 + ((lane >> 4) * 8);
        const int col   = n0 + ncol;
#pragma unroll
        for (int r = 0; r < 8; ++r)
            out[(size_t)(rbase + r) * D + col] = c[r];
    }
}

extern "C" void kernel_launch(void* const* d_in, const int* in_sizes, int n_in,
                              void* d_out, int out_size, void* d_ws, size_t ws_size,
                              hipStream_t stream) {
    const float*     x  = (const float*)d_in[0];
    const long long* ei = (const long long*)d_in[1];   // int64 [2, N_EDGES]
    const float*     Wm = (const float*)d_in[2];       // [64,64]
    const float*     Bm = (const float*)d_in[3];       // [64,64]
    float* out = (float*)d_out;

    float* agg = (float*)d_ws;                         // N_NODES * 64 floats
    float* deg = agg + (size_t)N_NODES * D;            // N_NODES floats

    // Phase 1: zero scratch
    int zn = N_NODES * D + N_NODES;
    zero_kernel<<<(zn + 255) / 256, 256, 0, stream>>>(agg, zn);

    // Phase 2: gather + scatter-add + degree
    long long total = (long long)N_EDGES * D;
    int blocks = (int)((total + 255) / 256);
    edge_scatter_kernel<<<blocks, 256, 0, stream>>>(
        ei, ei + N_EDGES, x, agg, deg, N_EDGES);

    // Phase 3: normalized dual GEMM via f32 WMMA
    int mtiles = (N_NODES + 15) / 16;                  // 6250
    gcn_wmma_kernel<<<(mtiles + 7) / 8, 256, 0, stream>>>(
        agg, deg, x, Wm, Bm, out, N_NODES);
}